// RippleNet_49855980372080
// MI455X (gfx1250) — compile-verified
//
#include <hip/hip_runtime.h>

// RippleNet on MI455X (gfx1250): relation-grouped WMMA f32 pipeline.
// N_ENT=200000, N_REL=200, D=64, N_HOP=2, N_MEM=16, B=4096

#define NENT 200000
#define NREL 200
#define DIM 64
#define NHOP 2
#define NMEM 16
#define BATCH 4096
#define PAIRS_PER_HOP (BATCH * NMEM)   // 65536
#define NPAIRS (NHOP * PAIRS_PER_HOP)  // 131072
#define NKEY (NHOP * NREL)             // 400

typedef float v2f __attribute__((ext_vector_type(2)));
typedef float v8f __attribute__((ext_vector_type(8)));

// ---------------------------------------------------------------------------
// K0: item_buf[b][d] = entity_emb[items[b]][d]; y_buf = 0; counts = 0
// ---------------------------------------------------------------------------
__global__ __launch_bounds__(256) void k_init(const int* __restrict__ items,
                                              const float* __restrict__ ent,
                                              float* __restrict__ item_buf,
                                              float* __restrict__ y_buf,
                                              int* __restrict__ counts) {
  int t = blockIdx.x * 256 + threadIdx.x;
  if (t < BATCH * DIM) {
    int b = t >> 6, d = t & 63;
    item_buf[t] = ent[items[b] * DIM + d];
    y_buf[t] = 0.0f;
  }
  if (t < NKEY) counts[t] = 0;
}

// ---------------------------------------------------------------------------
// K1: histogram pairs by (hop, relation)
// ---------------------------------------------------------------------------
__global__ __launch_bounds__(256) void k_count(const int* __restrict__ mem_r,
                                               int* __restrict__ counts) {
  int t = blockIdx.x * 256 + threadIdx.x;
  if (t < NPAIRS) {
    int hop = t >> 16;
    atomicAdd(&counts[hop * NREL + mem_r[t]], 1);
  }
}

// K1.5: 400-entry exclusive scan (tiny; single lane)
__global__ void k_scan(const int* __restrict__ counts, int* __restrict__ offsets,
                       int* __restrict__ cursor) {
  if (threadIdx.x == 0) {
    int acc = 0;
    for (int i = 0; i < NKEY; i++) {
      offsets[i] = acc;
      cursor[i] = acc;
      acc += counts[i];
    }
  }
}

// K1.75: scatter pair ids into relation segments
__global__ __launch_bounds__(256) void k_scatter(const int* __restrict__ mem_r,
                                                 int* __restrict__ cursor,
                                                 int* __restrict__ pairs) {
  int t = blockIdx.x * 256 + threadIdx.x;
  if (t < NPAIRS) {
    int hop = t >> 16;
    int slot = atomicAdd(&cursor[hop * NREL + mem_r[t]], 1);
    pairs[slot] = t;  // encodes hop*65536 + b*16 + m
  }
}

// ---------------------------------------------------------------------------
// K2: per relation segment, tiles of 16 pairs:
//     Rh(64x16) = R(64x64) @ H(64x16) via v_wmma_f32_16x16x4_f32,
//     logits[b,m] = Rh[:,n] . item[b_n]  (fused, Rh never stored)
// 128 threads = 4 waves; wave w owns output rows [16w, 16w+16).
// ---------------------------------------------------------------------------
__global__ __launch_bounds__(128) void k_rel_logits(
    int hop, const int* __restrict__ pairs, const int* __restrict__ counts,
    const int* __restrict__ offsets, const int* __restrict__ mem_h,
    const float* __restrict__ ent, const float* __restrict__ rel_emb,
    const float* __restrict__ item_buf, float* __restrict__ logits_buf) {
  __shared__ float Rl[64 * 66];   // padded stride: banks + b64 alignment
  __shared__ float Hl[16 * 68];   // H[col][k]
  __shared__ float Il[16 * 68];   // item rows per pair column
  __shared__ int pid[16];
  __shared__ float lt4[4][16];

  int rel = blockIdx.x;
  int key = hop * NREL + rel;
  int cnt = counts[key];
  int base = offsets[key];
  int ntile = (cnt + 15) >> 4;
  int t = threadIdx.x;

  // cooperative load of R into LDS (amortized over ~cnt/128 tiles per block)
  for (int i = t; i < 64 * 64; i += 128)
    Rl[(i >> 6) * 66 + (i & 63)] = rel_emb[rel * 4096 + i];

  int w = t >> 5;                 // wave id 0..3
  int L = t & 31;                 // lane
  int col = L & 15;               // output column / pair slot
  int khalf = (L < 16) ? 0 : 2;   // K split across lane halves (ISA A layout)

  for (int tile = blockIdx.y; tile < ntile; tile += gridDim.y) {
    __syncthreads();  // protect LDS reuse across iterations
    if (t < 16) {
      int j = tile * 16 + t;
      pid[t] = (j < cnt) ? pairs[base + j] : -1;
    }
    __syncthreads();

    // gather H rows (entity h vectors) and item rows: 8 threads per pair
    {
      int j = t >> 3;
      int d0 = t & 7;
      int p = pid[j];
      if (p >= 0) {
        int bm = p & 0xFFFF;
        int b = bm >> 4;
        const float* hsrc = ent + mem_h[p] * DIM;
        const float* isrc = item_buf + b * DIM;
#pragma unroll
        for (int k = 0; k < 8; k++) {
          int d = d0 + (k << 3);
          Hl[j * 68 + d] = hsrc[d];
          Il[j * 68 + d] = isrc[d];
        }
      } else {
#pragma unroll
        for (int k = 0; k < 8; k++) {
          int d = d0 + (k << 3);
          Hl[j * 68 + d] = 0.0f;
          Il[j * 68 + d] = 0.0f;
        }
      }
    }
    __syncthreads();

    // 16 chained f32 WMMAs: D(16x16) += A(16x4) * B(4x16) over K=64
    v8f c = {};
    int arow = w * 16 + col;
#pragma unroll
    for (int k0 = 0; k0 < 64; k0 += 4) {
      v2f a, bb;
      a.x = Rl[arow * 66 + k0 + khalf];
      a.y = Rl[arow * 66 + k0 + khalf + 1];
      bb.x = Hl[col * 68 + k0 + khalf];
      bb.y = Hl[col * 68 + k0 + khalf + 1];
      c = __builtin_amdgcn_wmma_f32_16x16x4_f32(false, a, false, bb, (short)0, c,
                                                false, false);
    }

    // fused logit: reduce c[r] * item over the 16 rows this wave owns
    float part = 0.0f;
    int rbase = w * 16 + ((L < 16) ? 0 : 8);
#pragma unroll
    for (int r = 0; r < 8; r++) part += c[r] * Il[col * 68 + rbase + r];
    part += __shfl_xor(part, 16, 32);  // combine lane halves (same col)
    if (L < 16) lt4[w][col] = part;    // staged: deterministic summation
    __syncthreads();

    if (t < 16 && pid[t] >= 0) {
      int bm = pid[t] & 0xFFFF;
      logits_buf[bm] = lt4[0][t] + lt4[1][t] + lt4[2][t] + lt4[3][t];
    }
  }
}

// ---------------------------------------------------------------------------
// K3: per 16-b tile: softmax over memory slots, o = sum p*t, y += o,
//     item = W @ (item + o) via WMMA, last hop: score = item . y
// 256 threads = 8 waves (waves 0..3 run the WMMA GEMM).
// ---------------------------------------------------------------------------
__global__ __launch_bounds__(256) void k_update(
    int hop, int last, const int* __restrict__ mem_t_idx,
    const float* __restrict__ ent, const float* __restrict__ Wmat,
    const float* __restrict__ logits_buf, float* __restrict__ item_buf,
    float* __restrict__ y_buf, float* __restrict__ out) {
  __shared__ float Wl[64 * 66];
  __shared__ float Xl[16 * 68];  // item (+o) columns
  __shared__ float Yl[16 * 68];  // running y per b
  __shared__ float Pl[16 * 17];  // softmax probs
  __shared__ int Tl[256];        // tail entity indices
  __shared__ float sc4[4][16];

  int b0 = blockIdx.x * 16;
  int t = threadIdx.x;

  for (int i = t; i < 64 * 64; i += 256) Wl[(i >> 6) * 66 + (i & 63)] = Wmat[i];
  for (int i = t; i < 16 * 64; i += 256) {
    int bl = i >> 6, d = i & 63;
    Xl[bl * 68 + d] = item_buf[(b0 + bl) * 64 + d];
  }
  Tl[t] = mem_t_idx[hop * PAIRS_PER_HOP + b0 * 16 + t];
  __syncthreads();

  // softmax: thread t = bl*16 + m; width-16 shfl reductions (wave32 halves)
  {
    int bl = t >> 4, m = t & 15;
    float l = logits_buf[b0 * 16 + t];
    float mx = l;
    for (int off = 8; off; off >>= 1) mx = fmaxf(mx, __shfl_xor(mx, off, 16));
    float e = __expf(l - mx);
    float s = e;
    for (int off = 8; off; off >>= 1) s += __shfl_xor(s, off, 16);
    Pl[bl * 17 + m] = e / s;
  }
  __syncthreads();

  // o + y accumulation: thread t -> b = t&15, d = (t>>4) + 16k
  {
    int bl = t & 15, dg = t >> 4;
#pragma unroll
    for (int k = 0; k < 4; k++) {
      int d = dg + k * 16;
      float acc = 0.0f;
#pragma unroll
      for (int m = 0; m < 16; m++)
        acc += Pl[bl * 17 + m] * ent[Tl[bl * 16 + m] * DIM + d];
      float ynew = y_buf[(b0 + bl) * 64 + d] + acc;
      y_buf[(b0 + bl) * 64 + d] = ynew;
      Yl[bl * 68 + d] = ynew;
      Xl[bl * 68 + d] += acc;  // item + o
    }
  }
  __syncthreads();

  // item_new(64x16) = W(64x64) @ X(64x16) via WMMA; waves 0..3
  int w = t >> 5, L = t & 31;
  if (w < 4) {
    int col = L & 15;
    int khalf = (L < 16) ? 0 : 2;
    int arow = w * 16 + col;
    v8f c = {};
#pragma unroll
    for (int k0 = 0; k0 < 64; k0 += 4) {
      v2f a, bb;
      a.x = Wl[arow * 66 + k0 + khalf];
      a.y = Wl[arow * 66 + k0 + khalf + 1];
      bb.x = Xl[col * 68 + k0 + khalf];
      bb.y = Xl[col * 68 + k0 + khalf + 1];
      c = __builtin_amdgcn_wmma_f32_16x16x4_f32(false, a, false, bb, (short)0, c,
                                                false, false);
    }
    int rbase = w * 16 + ((L < 16) ? 0 : 8);
    float part = 0.0f;
#pragma unroll
    for (int r = 0; r < 8; r++) {
      int row = rbase + r;
      item_buf[(b0 + col) * 64 + row] = c[r];
      part += c[r] * Yl[col * 68 + row];
    }
    if (last) {
      part += __shfl_xor(part, 16, 32);
      if (L < 16) sc4[w][col] = part;  // staged, deterministic
    }
  }
  __syncthreads();
  if (last && t < 16) out[b0 + t] = sc4[0][t] + sc4[1][t] + sc4[2][t] + sc4[3][t];
}

// ---------------------------------------------------------------------------
extern "C" void kernel_launch(void* const* d_in, const int* in_sizes, int n_in,
                              void* d_out, int out_size, void* d_ws,
                              size_t ws_size, hipStream_t stream) {
  const int* items = (const int*)d_in[0];
  const int* mem_h = (const int*)d_in[1];
  const int* mem_r = (const int*)d_in[2];
  const int* mem_t = (const int*)d_in[3];
  const float* ent = (const float*)d_in[4];
  const float* rel = (const float*)d_in[5];
  const float* Wm = (const float*)d_in[6];
  float* out = (float*)d_out;

  char* ws = (char*)d_ws;
  float* item_buf = (float*)ws;                              // 1 MB
  float* y_buf = (float*)(ws + (1u << 20));                  // 1 MB
  float* logits = (float*)(ws + (2u << 20));                 // 256 KB
  int* pairs = (int*)(ws + (2u << 20) + (256u << 10));       // 512 KB
  int* counts = (int*)(ws + (2u << 20) + (768u << 10));      // 400 ints
  int* offsets = counts + 512;
  int* cursor = offsets + 512;

  k_init<<<(BATCH * DIM + 255) / 256, 256, 0, stream>>>(items, ent, item_buf,
                                                        y_buf, counts);
  k_count<<<NPAIRS / 256, 256, 0, stream>>>(mem_r, counts);
  k_scan<<<1, 32, 0, stream>>>(counts, offsets, cursor);
  k_scatter<<<NPAIRS / 256, 256, 0, stream>>>(mem_r, cursor, pairs);

  for (int hop = 0; hop < NHOP; hop++) {
    k_rel_logits<<<dim3(NREL, 8), 128, 0, stream>>>(
        hop, pairs, counts, offsets, mem_h, ent, rel, item_buf, logits);
    k_update<<<BATCH / 16, 256, 0, stream>>>(hop, hop == NHOP - 1, mem_t, ent,
                                             Wm, logits, item_buf, y_buf, out);
  }
}